// topk_CE_68023692034065
// MI455X (gfx1250) — compile-verified
//
#include <hip/hip_runtime.h>
#include <math.h>

// ---------------------------------------------------------------------------
// OHEM BCE-with-logits, [64,1,512,512] fp32 input, int32 {0,1} target.
//
// out = (sum_b [white_sum_b + topk_black_sum_b]) / (sum_b [n_white_b + k_b]),
//   k_b = min(3*n_white_b, n_black_b)
//
// Fast path (expected for Bernoulli(1/2) targets): k == n_black -> the top-k
// negatives are ALL negatives, so one fused streaming pass suffices:
//   white: softplus(x) - x, black: softplus(x).
// Parallelized 64 samples x 8 slices = 512 blocks (16K wave32 waves) so the
// HBM stream (128 MB ~ 5.5 us @ 23.3 TB/s) actually saturates.
//
// Exact fallback (0 < k < n_black): per-sample 12/12/8-bit LDS-histogram
// radix select on the monotone float->uint key (softplus is monotone), with
// exact tie handling. Early-exits per sample when the fast path applied.
// ---------------------------------------------------------------------------

constexpr int BATCH   = 64;
constexpr int NPS     = 512 * 512;     // elements per sample
constexpr int SLICES  = 8;             // blocks per sample in pass A
constexpr int BLOCK   = 1024;          // 32 wave32 waves
constexpr int NV      = NPS / 4;       // float4 vectors per sample (65536)
constexpr int VPB     = NV / SLICES;   // vectors per pass-A block (8192)
constexpr int NB1     = 4096;          // radix bins, bits [31:20] / [19:8]
constexpr int NB3     = 256;           // bits [7:0]

// ---- workspace layout (floats unless noted) -------------------------------
// [0,512)      pW   : per-(sample,slice) white-loss partial
// [512,1024)   pB   : per-(sample,slice) all-black-loss partial
// [1024,1536)  pN   : (uint) per-(sample,slice) white count
// [1536,1600)  sampleSum
// [1600,1664)  sampleCnt
// [1664,1728)  sampleW  : per-sample white_sum (for fallback)
// [1728,1792)  flag (uint): 1 -> needs selection
// [1792,1856)  kval (uint)
#define WS_PW   0
#define WS_PB   512
#define WS_PN   1024
#define WS_SSUM 1536
#define WS_SCNT 1600
#define WS_SW   1664
#define WS_FLAG 1728
#define WS_KVAL 1792

__device__ __forceinline__ float softplusf(float x) {
  return fmaxf(x, 0.0f) + log1pf(expf(-fabsf(x)));
}
__device__ __forceinline__ unsigned f2ord(float x) {
  unsigned fu = __float_as_uint(x);
  return (fu & 0x80000000u) ? ~fu : (fu | 0x80000000u);
}
__device__ __forceinline__ float ord2f(unsigned u) {
  unsigned fu = (u & 0x80000000u) ? (u & 0x7FFFFFFFu) : ~u;
  return __uint_as_float(fu);
}

__device__ __forceinline__ float blockReduceSumF(float v, volatile float* scr) {
#pragma unroll
  for (int o = 16; o > 0; o >>= 1) v += __shfl_xor(v, o, 32);
  const int wave = threadIdx.x >> 5, lane = threadIdx.x & 31;
  __syncthreads();
  if (lane == 0) scr[wave] = v;
  __syncthreads();
  if (wave == 0) {
    float t = scr[lane];
#pragma unroll
    for (int o = 16; o > 0; o >>= 1) t += __shfl_xor(t, o, 32);
    if (lane == 0) scr[0] = t;
  }
  __syncthreads();
  float r = scr[0];
  __syncthreads();
  return r;
}
__device__ __forceinline__ unsigned blockReduceSumU(unsigned v, volatile unsigned* scr) {
#pragma unroll
  for (int o = 16; o > 0; o >>= 1) v += __shfl_xor(v, o, 32);
  const int wave = threadIdx.x >> 5, lane = threadIdx.x & 31;
  __syncthreads();
  if (lane == 0) scr[wave] = v;
  __syncthreads();
  if (wave == 0) {
    unsigned t = scr[lane];
#pragma unroll
    for (int o = 16; o > 0; o >>= 1) t += __shfl_xor(t, o, 32);
    if (lane == 0) scr[0] = t;
  }
  __syncthreads();
  unsigned r = scr[0];
  __syncthreads();
  return r;
}

// ---------------------------------------------------------------------------
// Pass A: fused streaming pass. 512 blocks, each owns 1/8 of one sample.
// ---------------------------------------------------------------------------
__global__ __launch_bounds__(BLOCK, 1) void ohem_passA(
    const float* __restrict__ x, const int* __restrict__ t, float* __restrict__ ws) {
  __shared__ float    redF[32];
  __shared__ unsigned redU[32];
  __shared__ int      asyncScratch[4096];  // 16 KB landing pad for async-to-LDS
                                           // prefetch of target; never read.
  const int b    = blockIdx.x >> 3;        // sample
  const int slot = blockIdx.x & 7;         // slice
  const int tid  = threadIdx.x;
  const int off  = slot * VPB;

  const float4* __restrict__ xb = (const float4*)(x + (size_t)b * NPS);
  const int4*   __restrict__ tb = (const int4*)(t + (size_t)b * NPS);
  const unsigned long long tBase = (unsigned long long)(t + (size_t)b * NPS);
  const unsigned ldsBase = (unsigned)(uintptr_t)asyncScratch; // LDS byte offset
  const unsigned ldsDst  = ldsBase + (unsigned)(tid & 1023) * 16u;

  float    wsum = 0.0f;   // positives: softplus(x) - x
  float    bsum = 0.0f;   // negatives: softplus(x), ALL of them
  unsigned cnt  = 0;      // n_white partial

  for (int i = tid; i < VPB; i += BLOCK) {
    // CDNA5 async engine: stage next iteration's target tile into LDS
    // (pure prefetch; ASYNCcnt-tracked; scratch never consumed).
    if (i + BLOCK < VPB) {
      unsigned gOff = (unsigned)((off + i + BLOCK) * 16);
      asm volatile("global_load_async_to_lds_b128 %0, %1, %2"
                   :: "v"(ldsDst), "v"(gOff), "s"(tBase) : "memory");
      __builtin_prefetch(xb + (off + i + BLOCK), 0, 1);  // global_prefetch_b8
    }
    float4 xv = xb[off + i];
    int4   tv = tb[off + i];
    const float xs[4] = {xv.x, xv.y, xv.z, xv.w};
    const int   ts[4] = {tv.x, tv.y, tv.z, tv.w};
#pragma unroll
    for (int j = 0; j < 4; ++j) {
      const float sp = softplusf(xs[j]);   // softplus(x)
      if (ts[j]) { cnt++; wsum += sp - xs[j]; }   // == softplus(-x)
      else       { bsum += sp; }
    }
  }
  asm volatile("s_wait_asynccnt 0x0" ::: "memory");
  __syncthreads();  // all async LDS writes retired before scratch reuse below

  const float    wTot = blockReduceSumF(wsum, redF);
  const float    bTot = blockReduceSumF(bsum, redF);
  const unsigned nTot = blockReduceSumU(cnt, redU);
  if (tid == 0) {
    ws[WS_PW + blockIdx.x] = wTot;
    ws[WS_PB + blockIdx.x] = bTot;
    ((unsigned*)ws)[WS_PN + blockIdx.x] = nTot;
  }
}

// ---------------------------------------------------------------------------
// Classify: fold 8 slices per sample (fixed order -> deterministic), decide
// fast vs fallback path per sample.
// ---------------------------------------------------------------------------
__global__ void ohem_classify(float* __restrict__ ws) {
  const int s = threadIdx.x;
  if (s >= BATCH) return;
  float wsum = 0.0f, ball = 0.0f;
  unsigned nw = 0;
  for (int j = 0; j < SLICES; ++j) {
    wsum += ws[WS_PW + s * SLICES + j];
    ball += ws[WS_PB + s * SLICES + j];
    nw   += ((unsigned*)ws)[WS_PN + s * SLICES + j];
  }
  const unsigned nb = (unsigned)NPS - nw;
  const unsigned k  = min(3u * nw, nb);
  ws[WS_SW + s]   = wsum;
  ws[WS_SCNT + s] = (float)(nw + k);
  ((unsigned*)ws)[WS_KVAL + s] = k;
  if (k == nb) {               // fast path: keep every negative
    ws[WS_SSUM + s] = wsum + ball;
    ((unsigned*)ws)[WS_FLAG + s] = 0u;
  } else if (k == 0u) {        // no negatives kept
    ws[WS_SSUM + s] = wsum;
    ((unsigned*)ws)[WS_FLAG + s] = 0u;
  } else {                     // needs exact radix select
    ws[WS_SSUM + s] = wsum;    // placeholder; select kernel overwrites
    ((unsigned*)ws)[WS_FLAG + s] = 1u;
  }
}

// ---------------------------------------------------------------------------
// Exact fallback: per-sample 12/12/8-bit radix select + top-k sum.
// Early-exits when the fast path already resolved the sample.
// ---------------------------------------------------------------------------
__global__ __launch_bounds__(BLOCK, 1) void ohem_select(
    const float* __restrict__ x, const int* __restrict__ t, float* __restrict__ ws) {
  __shared__ unsigned hist[NB1];
  __shared__ float    redF[32];
  __shared__ unsigned s_prefix;
  __shared__ unsigned s_rem;

  const int b = blockIdx.x;
  if (((const unsigned*)ws)[WS_FLAG + b] == 0u) return;   // fast path done

  const int tid = threadIdx.x;
  const unsigned k = ((const unsigned*)ws)[WS_KVAL + b];  // 0 < k < n_black
  const float4* __restrict__ xb = (const float4*)(x + (size_t)b * NPS);
  const int4*   __restrict__ tb = (const int4*)(t + (size_t)b * NPS);

  // ---- level 1: histogram of bits [31:20] over negatives ------------------
  for (int i = tid; i < NB1; i += BLOCK) hist[i] = 0u;
  __syncthreads();
  for (int i = tid; i < NV; i += BLOCK) {
    __builtin_prefetch(xb + i + BLOCK, 0, 1);
    float4 xv = xb[i];
    int4   tv = tb[i];
    const float xs[4] = {xv.x, xv.y, xv.z, xv.w};
    const int   ts[4] = {tv.x, tv.y, tv.z, tv.w};
#pragma unroll
    for (int j = 0; j < 4; ++j)
      if (!ts[j]) atomicAdd(&hist[f2ord(xs[j]) >> 20], 1u);
  }
  __syncthreads();
  if (tid == 0) {
    unsigned rem = k, cum = 0, sel = 0;
    for (int j = NB1 - 1; j >= 0; --j) {
      unsigned c = hist[j];
      if (cum + c >= rem) { sel = (unsigned)j; rem -= cum; break; }
      cum += c;
    }
    s_prefix = sel; s_rem = rem;
  }
  __syncthreads();

  // ---- level 2: bits [19:8] within prefix ----------------------------------
  const unsigned p1 = s_prefix;
  __syncthreads();
  for (int i = tid; i < NB1; i += BLOCK) hist[i] = 0u;
  __syncthreads();
  for (int i = tid; i < NV; i += BLOCK) {
    float4 xv = xb[i];
    int4   tv = tb[i];
    const float xs[4] = {xv.x, xv.y, xv.z, xv.w};
    const int   ts[4] = {tv.x, tv.y, tv.z, tv.w};
#pragma unroll
    for (int j = 0; j < 4; ++j) {
      if (!ts[j]) {
        unsigned u = f2ord(xs[j]);
        if ((u >> 20) == p1) atomicAdd(&hist[(u >> 8) & 0xFFFu], 1u);
      }
    }
  }
  __syncthreads();
  if (tid == 0) {
    unsigned rem = s_rem, cum = 0, sel = 0;
    for (int j = NB1 - 1; j >= 0; --j) {
      unsigned c = hist[j];
      if (cum + c >= rem) { sel = (unsigned)j; rem -= cum; break; }
      cum += c;
    }
    s_prefix = (p1 << 12) | sel; s_rem = rem;
  }
  __syncthreads();

  // ---- level 3: bits [7:0] within prefix -----------------------------------
  const unsigned p2 = s_prefix;
  __syncthreads();
  for (int i = tid; i < NB3; i += BLOCK) hist[i] = 0u;
  __syncthreads();
  for (int i = tid; i < NV; i += BLOCK) {
    float4 xv = xb[i];
    int4   tv = tb[i];
    const float xs[4] = {xv.x, xv.y, xv.z, xv.w};
    const int   ts[4] = {tv.x, tv.y, tv.z, tv.w};
#pragma unroll
    for (int j = 0; j < 4; ++j) {
      if (!ts[j]) {
        unsigned u = f2ord(xs[j]);
        if ((u >> 8) == p2) atomicAdd(&hist[u & 0xFFu], 1u);
      }
    }
  }
  __syncthreads();
  if (tid == 0) {
    unsigned rem = s_rem, cum = 0, sel = 0;
    for (int j = NB3 - 1; j >= 0; --j) {
      unsigned c = hist[j];
      if (cum + c >= rem) { sel = (unsigned)j; rem -= cum; break; }
      cum += c;
    }
    s_prefix = (p2 << 8) | sel;   // exact 32-bit threshold key
    s_rem    = rem;               // ties at threshold to include
  }
  __syncthreads();

  // ---- final: sum losses strictly above threshold --------------------------
  const unsigned uthr = s_prefix;
  float bsum = 0.0f;
  for (int i = tid; i < NV; i += BLOCK) {
    __builtin_prefetch(xb + i + BLOCK, 0, 1);
    float4 xv = xb[i];
    int4   tv = tb[i];
    const float xs[4] = {xv.x, xv.y, xv.z, xv.w};
    const int   ts[4] = {tv.x, tv.y, tv.z, tv.w};
#pragma unroll
    for (int j = 0; j < 4; ++j)
      if (!ts[j] && f2ord(xs[j]) > uthr) bsum += softplusf(xs[j]);
  }
  float blackSum = blockReduceSumF(bsum, redF);
  if (tid == 0) {
    blackSum += (float)s_rem * softplusf(ord2f(uthr));   // exact ties
    ws[WS_SSUM + b] = ws[WS_SW + b] + blackSum;
  }
}

__global__ void ohem_finalize(const float* __restrict__ ws, float* __restrict__ out) {
  if (threadIdx.x == 0) {
    float ts = 0.0f, tc = 0.0f;
    for (int i = 0; i < BATCH; ++i) { ts += ws[WS_SSUM + i]; tc += ws[WS_SCNT + i]; }
    out[0] = ts / tc;
  }
}

extern "C" void kernel_launch(void* const* d_in, const int* in_sizes, int n_in,
                              void* d_out, int out_size, void* d_ws, size_t ws_size,
                              hipStream_t stream) {
  (void)in_sizes; (void)n_in; (void)out_size; (void)ws_size;
  const float* x = (const float*)d_in[0];
  const int*   t = (const int*)d_in[1];
  float* ws = (float*)d_ws;

  ohem_passA   <<<BATCH * SLICES, BLOCK, 0, stream>>>(x, t, ws);
  ohem_classify<<<1, 64, 0, stream>>>(ws);
  ohem_select  <<<BATCH, BLOCK, 0, stream>>>(x, t, ws);
  ohem_finalize<<<1, 64, 0, stream>>>(ws, (float*)d_out);
}